// MPNNBlock_44384192037434
// MI455X (gfx1250) — compile-verified
//
#include <hip/hip_runtime.h>
#include <cstdint>
#include <cstddef>

// ---------------------------------------------------------------------------
// GAT block (GATConv H=2, C=64, edge features, self-loops w/ mean fill,
// segment softmax, head-mean + bias, LayerNorm, residual) for MI455X gfx1250.
//
// Strategy:
//  * algebraically collapse ea@W_e and the att_* reductions into 64-vectors
//  * single fused pass over edge_attr (a_e dots + column-sum for loop_attr)
//  * fp32 WMMA 16x16x4 for xp = x @ W (the only GEMM whose output is needed)
//  * segment softmax via ordered-uint atomicMax + float atomicAdd (L2-resident)
//  * one wave per edge for the 128-channel gather/scatter (float4 + atomics)
// ---------------------------------------------------------------------------

typedef __attribute__((ext_vector_type(2))) float v2f;
typedef __attribute__((ext_vector_type(8))) float v8f;

#define LEAKY 0.2f
#define LN_EPS 1e-5f

__device__ __forceinline__ unsigned ordf(float f) {
  unsigned b = __float_as_uint(f);
  return (b & 0x80000000u) ? ~b : (b | 0x80000000u);
}
__device__ __forceinline__ float unordf(unsigned u) {
  unsigned b = (u & 0x80000000u) ? (u & 0x7fffffffu) : ~u;
  return __uint_as_float(b);
}

// ---------------------------------------------------------------------------
// Kernel A: reduce W/W_e against att_{src,dst,edge} into H*D vectors.
//   v_src[h*64+d] = sum_c W[d*128 + h*64 + c] * att_src[h*64+c]   (ditto dst,e)
// Also zeroes colsum[64] and loop_ae[2]. Single block.
// ---------------------------------------------------------------------------
__global__ __launch_bounds__(256) void prep_vecs(
    const float* __restrict__ W, const float* __restrict__ We,
    const float* __restrict__ att_src, const float* __restrict__ att_dst,
    const float* __restrict__ att_edge, float* __restrict__ ws) {
  int t = threadIdx.x;
  if (t < 66) ws[384 + t] = 0.f;  // colsum[64] + loop_ae[2]
  if (t < 128) {
    int h = t >> 6, d = t & 63;
    float s1 = 0.f, s2 = 0.f, s3 = 0.f;
    for (int c = 0; c < 64; ++c) {
      float w  = W [d * 128 + h * 64 + c];
      float we = We[d * 128 + h * 64 + c];
      s1 += w  * att_src [h * 64 + c];
      s2 += w  * att_dst [h * 64 + c];
      s3 += we * att_edge[h * 64 + c];
    }
    ws[0   + t] = s1;   // v_src
    ws[128 + t] = s2;   // v_dst
    ws[256 + t] = s3;   // v_e
  }
}

// ---------------------------------------------------------------------------
// Kernel B: xp = x @ W via V_WMMA_F32_16X16X4_F32.
// Block = 256 threads = 8 waves. Block computes a 16-row x 128-col tile;
// wave w owns output columns [16w, 16w+16). K=64 -> 16 WMMA steps of K=4.
// W (64x128, 32KB) + x tile (16x64, 4KB) staged in LDS.
// ---------------------------------------------------------------------------
__global__ __launch_bounds__(256) void gemm_xp(
    const float* __restrict__ x, const float* __restrict__ W,
    float* __restrict__ xp, int nrows) {
  __shared__ float sA[16 * 64];
  __shared__ float sB[64 * 128];
  const int tid  = threadIdx.x;
  const int row0 = blockIdx.x * 16;

  for (int i = tid; i < 64 * 128; i += 256) sB[i] = W[i];
  for (int i = tid; i < 16 * 64; i += 256) {
    int r = i >> 6, c = i & 63;
    int gr = row0 + r;
    sA[i] = (gr < nrows) ? x[(size_t)gr * 64 + c] : 0.f;
  }
  __syncthreads();

  const int wave  = tid >> 5;
  const int lane  = tid & 31;
  const int colb  = wave * 16;
  const int m     = lane & 15;            // row (A) / col (B,C) within tile
  const int khalf = (lane >> 4) * 2;      // 0 for lanes 0-15, 2 for 16-31

  v8f acc = {};
#pragma unroll
  for (int k0 = 0; k0 < 64; k0 += 4) {
    v2f a, b;
    // A 16x4 fp32 layout: VGPR0 = {K0 | K2}, VGPR1 = {K1 | K3} across lane halves
    a.x = sA[m * 64 + k0 + khalf];
    a.y = sA[m * 64 + k0 + khalf + 1];
    // B 4x16: rows striped across lane halves / VGPRs analogously
    b.x = sB[(k0 + khalf)     * 128 + colb + m];
    b.y = sB[(k0 + khalf + 1) * 128 + colb + m];
    acc = __builtin_amdgcn_wmma_f32_16x16x4_f32(
        /*neg_a=*/false, a, /*neg_b=*/false, b,
        /*c_mod=*/(short)0, acc, /*reuse_a=*/false, /*reuse_b=*/false);
  }

  // C/D 16x16 f32 layout: VGPR j -> M = j + 8*(lane/16), N = lane%16
  const int nCol  = colb + m;
  const int mBase = (lane >> 4) * 8;
#pragma unroll
  for (int j = 0; j < 8; ++j) {
    int r = row0 + mBase + j;
    if (r < nrows) xp[(size_t)r * 128 + nCol] = acc[j];
  }
}

// ---------------------------------------------------------------------------
// Kernel C: per-node a_src/a_dst = x[n] . v_{src,dst}[h]; zero segment state
// and the output accumulator row.
// ---------------------------------------------------------------------------
__global__ __launch_bounds__(256) void node_init(
    const float* __restrict__ x, const float* __restrict__ vvec,
    float* __restrict__ a_src, float* __restrict__ a_dst,
    unsigned* __restrict__ seg_max, float* __restrict__ seg_sum,
    float* __restrict__ outacc, int N) {
  __shared__ float sV[256];                 // v_src(128) | v_dst(128)
  int tid = threadIdx.x;
  sV[tid] = vvec[tid];
  __syncthreads();
  int n = blockIdx.x * 256 + tid;
  if (n >= N) return;

  const float4* row = (const float4*)(x + (size_t)n * 64);
  float s0 = 0.f, s1 = 0.f, t0 = 0.f, t1 = 0.f;
#pragma unroll
  for (int i = 0; i < 16; ++i) {
    float4 v = row[i];
    int c = 4 * i;
    s0 += v.x * sV[c]       + v.y * sV[c + 1]       + v.z * sV[c + 2]       + v.w * sV[c + 3];
    s1 += v.x * sV[64 + c]  + v.y * sV[64 + c + 1]  + v.z * sV[64 + c + 2]  + v.w * sV[64 + c + 3];
    t0 += v.x * sV[128 + c] + v.y * sV[128 + c + 1] + v.z * sV[128 + c + 2] + v.w * sV[128 + c + 3];
    t1 += v.x * sV[192 + c] + v.y * sV[192 + c + 1] + v.z * sV[192 + c + 2] + v.w * sV[192 + c + 3];
  }
  a_src[2 * n] = s0;  a_src[2 * n + 1] = s1;
  a_dst[2 * n] = t0;  a_dst[2 * n + 1] = t1;
  seg_max[2 * n] = 0u; seg_max[2 * n + 1] = 0u;   // ordered-uint -inf sentinel
  seg_sum[2 * n] = 0.f; seg_sum[2 * n + 1] = 0.f;

  float4 z = make_float4(0.f, 0.f, 0.f, 0.f);
  float4* oa = (float4*)(outacc + (size_t)n * 128);
#pragma unroll
  for (int i = 0; i < 32; ++i) oa[i] = z;
}

// ---------------------------------------------------------------------------
// Kernel D: single pass over edge_attr. Per real edge: a_e[e,h] = ea[e].v_e[h].
// Fused column-sum (for the self-loop mean attr) via LDS ds_add_f32, one
// global atomicAdd per column per block.
// ---------------------------------------------------------------------------
__global__ __launch_bounds__(256) void edge_ae(
    const float* __restrict__ ea, const float* __restrict__ v_e,
    float* __restrict__ colsum, float* __restrict__ aeE, int E) {
  __shared__ float sCol[64];
  __shared__ float sV[128];
  int tid = threadIdx.x;
  if (tid < 64) sCol[tid] = 0.f;
  if (tid < 128) sV[tid] = v_e[tid];
  __syncthreads();

  int e = blockIdx.x * 256 + tid;
  if (e < E) {
    const float4* row = (const float4*)(ea + (size_t)e * 64);
    float d0 = 0.f, d1 = 0.f;
#pragma unroll
    for (int i = 0; i < 16; ++i) {
      float4 v = row[i];
      int c = 4 * i;
      d0 += v.x * sV[c]      + v.y * sV[c + 1]      + v.z * sV[c + 2]      + v.w * sV[c + 3];
      d1 += v.x * sV[64 + c] + v.y * sV[64 + c + 1] + v.z * sV[64 + c + 2] + v.w * sV[64 + c + 3];
      atomicAdd(&sCol[c],     v.x);
      atomicAdd(&sCol[c + 1], v.y);
      atomicAdd(&sCol[c + 2], v.z);
      atomicAdd(&sCol[c + 3], v.w);
    }
    aeE[2 * e] = d0;
    aeE[2 * e + 1] = d1;
  }
  __syncthreads();
  if (tid < 64) atomicAdd(&colsum[tid], sCol[tid]);
}

// Kernel D2: a_e of every self-loop edge is the same scalar per head:
//   loop_ae[h] = (colsum/E) . v_e[h]
__global__ void loop_ae_k(float* __restrict__ ws, int E) {
  int t = threadIdx.x;
  if (t < 2) {
    float invE = 1.f / (float)E;
    float s = 0.f;
    for (int c = 0; c < 64; ++c)
      s += ws[384 + c] * invE * ws[256 + t * 64 + c];
    ws[448 + t] = s;
  }
}

// ---------------------------------------------------------------------------
// Kernel E: alpha = leaky_relu(a_src[src]+a_dst[dst]+a_e); segment max (dst).
// ---------------------------------------------------------------------------
__global__ __launch_bounds__(256) void edge_alpha(
    const int* __restrict__ ei, const float* __restrict__ a_src,
    const float* __restrict__ a_dst, const float* __restrict__ aeE,
    const float* __restrict__ loop_ae, float* __restrict__ p,
    unsigned* __restrict__ seg_max, int E, int N) {
  int e = blockIdx.x * 256 + threadIdx.x;
  if (e >= E + N) return;
  int s, d; float ae0, ae1;
  if (e < E) {
    s = ei[e]; d = ei[E + e];
    ae0 = aeE[2 * e]; ae1 = aeE[2 * e + 1];
  } else {
    s = d = e - E;
    ae0 = loop_ae[0]; ae1 = loop_ae[1];
  }
  float al0 = a_src[2 * s]     + a_dst[2 * d]     + ae0;
  float al1 = a_src[2 * s + 1] + a_dst[2 * d + 1] + ae1;
  al0 = (al0 > 0.f) ? al0 : LEAKY * al0;
  al1 = (al1 > 0.f) ? al1 : LEAKY * al1;
  p[2 * e] = al0;
  p[2 * e + 1] = al1;
  atomicMax(&seg_max[2 * d],     ordf(al0));
  atomicMax(&seg_max[2 * d + 1], ordf(al1));
}

// Kernel F: p = exp(alpha - m[dst]); segment sum (dst).  (p overwrites alpha)
__global__ __launch_bounds__(256) void edge_p(
    const int* __restrict__ ei, float* __restrict__ p,
    const unsigned* __restrict__ seg_max, float* __restrict__ seg_sum,
    int E, int N) {
  int e = blockIdx.x * 256 + threadIdx.x;
  if (e >= E + N) return;
  int d = (e < E) ? ei[E + e] : (e - E);
#pragma unroll
  for (int h = 0; h < 2; ++h) {
    float m  = unordf(seg_max[2 * d + h]);
    float pv = expf(p[2 * e + h] - m);
    p[2 * e + h] = pv;
    atomicAdd(&seg_sum[2 * d + h], pv);
  }
}

// ---------------------------------------------------------------------------
// Kernel G: scatter msg = xp[src,h,:] * att into outacc[dst,h,:].
// One wave per edge; lane owns 4 consecutive channels of the 128 (H*C).
// xp + outacc (51 MB each) are L2-resident on MI455X (192 MB L2).
// ---------------------------------------------------------------------------
__global__ __launch_bounds__(256) void scatter_msg(
    const int* __restrict__ ei, const float* __restrict__ p,
    const float* __restrict__ seg_sum, const float* __restrict__ xp,
    float* __restrict__ outacc, int E, int N) {
  int gid  = blockIdx.x * 256 + threadIdx.x;
  int e    = gid >> 5;
  int lane = gid & 31;
  if (e >= E + N) return;
  int s, d;
  if (e < E) { s = ei[e]; d = ei[E + e]; }
  else       { s = d = e - E; }
  int h = lane >> 4;                        // channels [0,64)->h0, [64,128)->h1
  float att = p[2 * e + h] / (seg_sum[2 * d + h] + 1e-16f);
  float4 xv = *(const float4*)(xp + (size_t)s * 128 + lane * 4);
  float* o = outacc + (size_t)d * 128 + lane * 4;
  atomicAdd(o + 0, xv.x * att);
  atomicAdd(o + 1, xv.y * att);
  atomicAdd(o + 2, xv.z * att);
  atomicAdd(o + 3, xv.w * att);
}

// ---------------------------------------------------------------------------
// Kernel H: head mean + bias, LayerNorm, residual.
// ---------------------------------------------------------------------------
__global__ __launch_bounds__(256) void finalize(
    const float* __restrict__ x, const float* __restrict__ outacc,
    const float* __restrict__ bias, const float* __restrict__ gamma,
    const float* __restrict__ beta, float* __restrict__ out, int N) {
  int n = blockIdx.x * 256 + threadIdx.x;
  if (n >= N) return;
  const float* a = outacc + (size_t)n * 128;
  float o[64];
  float mu = 0.f;
#pragma unroll
  for (int c = 0; c < 64; ++c) {
    float v = 0.5f * (a[c] + a[64 + c]) + bias[c];
    o[c] = v;
    mu += v;
  }
  mu *= (1.f / 64.f);
  float var = 0.f;
#pragma unroll
  for (int c = 0; c < 64; ++c) {
    float dlt = o[c] - mu;
    var += dlt * dlt;
  }
  var *= (1.f / 64.f);
  float inv = rsqrtf(var + LN_EPS);
#pragma unroll
  for (int c = 0; c < 64; ++c) {
    out[(size_t)n * 64 + c] =
        x[(size_t)n * 64 + c] + (o[c] - mu) * inv * gamma[c] + beta[c];
  }
}

// ---------------------------------------------------------------------------
extern "C" void kernel_launch(void* const* d_in, const int* in_sizes, int n_in,
                              void* d_out, int out_size, void* d_ws, size_t ws_size,
                              hipStream_t stream) {
  (void)n_in; (void)out_size; (void)ws_size;
  const float* x        = (const float*)d_in[0];
  const int*   ei       = (const int*)  d_in[1];   // [2,E] row-major: src | dst
  const float* ea       = (const float*)d_in[2];
  const float* W        = (const float*)d_in[3];
  const float* We       = (const float*)d_in[4];
  const float* att_src  = (const float*)d_in[5];
  const float* att_dst  = (const float*)d_in[6];
  const float* att_edge = (const float*)d_in[7];
  const float* bias     = (const float*)d_in[8];
  const float* gamma    = (const float*)d_in[9];
  const float* beta     = (const float*)d_in[10];
  float* out = (float*)d_out;

  const int N = in_sizes[0] / 64;
  const int E = in_sizes[1] / 2;
  const int Etot = E + N;

  // Workspace layout (floats)
  float* ws = (float*)d_ws;
  //   [0,128)   v_src | [128,256) v_dst | [256,384) v_e
  //   [384,448) colsum | [448,450) loop_ae
  size_t off = 512;
  float*    a_src   = ws + off;              off += (size_t)2 * N;
  float*    a_dst   = ws + off;              off += (size_t)2 * N;
  unsigned* seg_max = (unsigned*)(ws + off); off += (size_t)2 * N;
  float*    seg_sum = ws + off;              off += (size_t)2 * N;
  float*    p       = ws + off;              off += (size_t)2 * Etot;  // alpha -> p
  float*    aeE     = ws + off;              off += (size_t)2 * E;
  off = (off + 63) & ~(size_t)63;
  float*    xp      = ws + off;              off += (size_t)N * 128;
  float*    outacc  = ws + off;              off += (size_t)N * 128;

  prep_vecs<<<1, 256, 0, stream>>>(W, We, att_src, att_dst, att_edge, ws);
  gemm_xp<<<(N + 15) / 16, 256, 0, stream>>>(x, W, xp, N);
  node_init<<<(N + 255) / 256, 256, 0, stream>>>(x, ws, a_src, a_dst,
                                                 seg_max, seg_sum, outacc, N);
  edge_ae<<<(E + 255) / 256, 256, 0, stream>>>(ea, ws + 256, ws + 384, aeE, E);
  loop_ae_k<<<1, 32, 0, stream>>>(ws, E);
  edge_alpha<<<(Etot + 255) / 256, 256, 0, stream>>>(ei, a_src, a_dst, aeE,
                                                     ws + 448, p, seg_max, E, N);
  edge_p<<<(Etot + 255) / 256, 256, 0, stream>>>(ei, p, seg_max, seg_sum, E, N);
  {
    long long threads = (long long)Etot * 32;
    int blocks = (int)((threads + 255) / 256);
    scatter_msg<<<blocks, 256, 0, stream>>>(ei, p, seg_sum, xp, outacc, E, N);
  }
  finalize<<<(N + 255) / 256, 256, 0, stream>>>(x, outacc, bias, gamma, beta, out, N);
}